// DilationConv_52261162058289
// MI455X (gfx1250) — compile-verified
//
#include <hip/hip_runtime.h>
#include <hip/hip_bf16.h>
#include <stdint.h>
#include <stddef.h>

// ---------------- problem constants (from reference) ----------------
#define Ndim 16
#define Cdim 64
#define Hdim 256
#define Wdim 256
#define CVALF (-10000.0f)

// ---------------- tiling ----------------
#define ROWS   16              // output rows per block
#define TROWS  (ROWS + 2)      // input rows staged in LDS
#define LSTRIDE 260            // floats per LDS row: 1 halo + 256 + 1 halo + 2 pad
#define LSHIFT  3              // leading floats so interior (+1) lands 16B-aligned
#define LDS_FLOATS (LSHIFT + TROWS * LSTRIDE)

typedef float vfloat4 __attribute__((ext_vector_type(4)));
typedef int   v4i     __attribute__((__vector_size__(4 * sizeof(int))));
typedef __attribute__((address_space(1))) v4i gas_v4i;   // __device__ int4
typedef __attribute__((address_space(3))) v4i las_v4i;   // __shared__ int4

// ---------------- CDNA5 async global->LDS copy ----------------
#if defined(__has_builtin)
# if __has_builtin(__builtin_amdgcn_global_load_async_to_lds_b128)
#  define USE_ASYNC_BUILTIN 1
# endif
# if __has_builtin(__builtin_amdgcn_s_wait_asynccnt)
#  define USE_WAIT_BUILTIN 1
# endif
#endif

__device__ __forceinline__ void async_ld_b128(const float* gp, const float* lp) {
#if defined(USE_ASYNC_BUILTIN)
    // Signature (from clang diagnostic): param0 = int4 __device__* (global src),
    // param1 = LDS dst, then imm offset, imm cpol. Build addrspace pointers via
    // integer casts: flat-shared addr[31:0] == LDS byte offset (ISA §10.2).
    __builtin_amdgcn_global_load_async_to_lds_b128(
        (gas_v4i*)(uintptr_t)gp,
        (las_v4i*)(uint32_t)(uintptr_t)lp,
        0, 0);
#else
    uint32_t loff = (uint32_t)(uintptr_t)lp;          // wave-relative LDS byte addr
    uint64_t ga   = (uint64_t)(uintptr_t)gp;
    asm volatile("global_load_async_to_lds_b128 %0, %1, off"
                 :: "v"(loff), "v"(ga) : "memory");
#endif
}

__device__ __forceinline__ void async_wait0() {
#if defined(USE_WAIT_BUILTIN)
    __builtin_amdgcn_s_wait_asynccnt(0);
#else
    asm volatile("s_wait_asynccnt 0x0" ::: "memory");
#endif
}

// ---------------- kernel ----------------
__global__ __launch_bounds__(256)
void dilate3x3_kernel(const float* __restrict__ x,
                      const float* __restrict__ se,
                      float* __restrict__ out)
{
    __shared__ float lds[LDS_FLOATS];

    const int tid  = threadIdx.x;
    const int tile = blockIdx.x;            // row band: 0..Hdim/ROWS-1
    const int c    = blockIdx.y;
    const int n    = blockIdx.z;
    const int h0   = tile * ROWS;           // first output row of this band

    const float* __restrict__ xc = x   + (size_t)(n * Cdim + c) * (Hdim * Wdim);
    float*       __restrict__ oc = out + (size_t)(n * Cdim + c) * (Hdim * Wdim);

    // ---- prefill halo columns (and out-of-image rows) with CVAL ----
    for (int r = tid; r < TROWS; r += 256) {
        const int base = LSHIFT + r * LSTRIDE;
        lds[base]            = CVALF;       // padded col -1
        lds[base + 1 + Wdim] = CVALF;       // padded col W
    }
    if (h0 == 0) {                          // LDS row 0 == global row -1
        for (int w = tid; w < Wdim; w += 256) lds[LSHIFT + 1 + w] = CVALF;
    }
    if (h0 + ROWS == Hdim) {                // last LDS row == global row H
        const int base = LSHIFT + (TROWS - 1) * LSTRIDE + 1;
        for (int w = tid; w < Wdim; w += 256) lds[base + w] = CVALF;
    }

    // ---- async-copy the in-image interior rows into LDS (16B per lane) ----
    const int CHUNKS = TROWS * (Wdim / 4);  // 18 * 64 = 1152 float4 chunks
    for (int idx = tid; idx < CHUNKS; idx += 256) {
        const int r  = idx >> 6;            // LDS row
        const int cx = idx & 63;            // float4 chunk in row
        const int g  = h0 - 1 + r;          // global image row
        if ((unsigned)g < (unsigned)Hdim) {
            const float* gp = xc + (size_t)g * Wdim + cx * 4;
            const float* lp = &lds[LSHIFT + r * LSTRIDE + 1 + cx * 4]; // 16B aligned
            async_ld_b128(gp, lp);
        }
    }
    async_wait0();
    __syncthreads();

    // ---- structuring element for this channel (block-uniform -> SGPRs) ----
    const float* sp = se + c * 9;
    float s[3][3];
#pragma unroll
    for (int i = 0; i < 3; ++i)
#pragma unroll
        for (int j = 0; j < 3; ++j)
            s[i][j] = sp[i * 3 + j];

    // ---- compute: each thread -> 4 rows x 4 cols of output ----
    const int cx = tid & 63;                // column chunk (4 outputs wide)
    const int r0 = tid >> 6;                // 0..3
#pragma unroll
    for (int rr = 0; rr < ROWS; rr += 4) {
        const int r = r0 + rr;              // output row within band
        float acc0 = 0.f, acc1 = 0.f, acc2 = 0.f, acc3 = 0.f;
#pragma unroll
        for (int i = 0; i < 3; ++i) {
            // window cols (padded) [4cx-1 .. 4cx+4] == lds row base + 4cx + k
            const float* row = &lds[LSHIFT + (r + i) * LSTRIDE + cx * 4];
            const float w0 = row[0], w1 = row[1], w2 = row[2],
                        w3 = row[3], w4 = row[4], w5 = row[5];
            // fmaxf chains -> v_max3_f32 fusion in the backend
            const float c0 = fmaxf(fmaxf(w0 + s[i][0], w1 + s[i][1]), w2 + s[i][2]);
            const float c1 = fmaxf(fmaxf(w1 + s[i][0], w2 + s[i][1]), w3 + s[i][2]);
            const float c2 = fmaxf(fmaxf(w2 + s[i][0], w3 + s[i][1]), w4 + s[i][2]);
            const float c3 = fmaxf(fmaxf(w3 + s[i][0], w4 + s[i][1]), w5 + s[i][2]);
            if (i == 0) { acc0 = c0; acc1 = c1; acc2 = c2; acc3 = c3; }
            else {
                acc0 = fmaxf(acc0, c0); acc1 = fmaxf(acc1, c1);
                acc2 = fmaxf(acc2, c2); acc3 = fmaxf(acc3, c3);
            }
        }
        vfloat4 v; v.x = acc0; v.y = acc1; v.z = acc2; v.w = acc3;
        vfloat4* dst = (vfloat4*)(oc + (size_t)(h0 + r) * Wdim + cx * 4);
        __builtin_nontemporal_store(v, dst);  // global_store_b128 th:NT (stream out)
    }
}

// ---------------- launch ----------------
extern "C" void kernel_launch(void* const* d_in, const int* in_sizes, int n_in,
                              void* d_out, int out_size, void* d_ws, size_t ws_size,
                              hipStream_t stream) {
    const float* x  = (const float*)d_in[0];   // (16,64,256,256) f32
    const float* se = (const float*)d_in[1];   // (64,3,3) f32
    float* out = (float*)d_out;                // (16,64,256,256) f32

    dim3 grid(Hdim / ROWS, Cdim, Ndim);        // 16 x 64 x 16 = 16384 blocks
    dilate3x3_kernel<<<grid, 256, 0, stream>>>(x, se, out);
}